// TC_MoE_88596585382782
// MI455X (gfx1250) — compile-verified
//
#include <hip/hip_runtime.h>

typedef __attribute__((ext_vector_type(2))) float v2f;
typedef __attribute__((ext_vector_type(8))) float v8f;

constexpr int DIM = 1024;
constexpr int HID = 4096;
constexpr int NT  = 2048;
constexpr int NE  = 8;
// total gathered rows: N*k token rows + per-expert sentinel, each expert
// segment padded to a multiple of 64 -> <= N*k + NE*64
constexpr int MAXROWS = NT * 2 + NE * 64;   // 4608

// ---------------- workspace layout (bytes) ----------------
constexpr size_t OFF_CNT    = 0;                          // int cnt_se[16]
constexpr size_t OFF_BASE   = 64;                         // int base[9]
constexpr size_t OFF_FILL   = 128;                        // int fill[8]
constexpr size_t OFF_TOPIDX = 192;                        // int topidx[NT*2]
constexpr size_t OFF_TOPW   = OFF_TOPIDX + NT * 2 * 4;    // float topw[NT*2]
constexpr size_t OFF_LOC    = OFF_TOPW   + NT * 2 * 4;    // int loc[NT*2]
constexpr size_t OFF_ROWTOK = OFF_LOC    + NT * 2 * 4;    // int rowtok[MAXROWS]
constexpr size_t OFF_CS     = OFF_ROWTOK + MAXROWS * 4;   // float Cs[2*DIM]
constexpr size_t OFF_H      = ((OFF_CS + 2 * DIM * 4 + 255) / 256) * 256; // float h[MAXROWS*HID]
constexpr size_t OFF_Y      = OFF_H + (size_t)MAXROWS * HID * 4;          // float y[MAXROWS*DIM]

// ---------------- init: zero counters, rowtok = -1 ----------------
__global__ void init_kernel(int* cnt_se, int* fill, int* rowtok) {
    int i = blockIdx.x * 256 + threadIdx.x;
    if (i < MAXROWS) rowtok[i] = -1;
    if (blockIdx.x == 0) {
        if (threadIdx.x < 16) cnt_se[threadIdx.x] = 0;
        if (threadIdx.x < 8)  fill[threadIdx.x]  = 0;
    }
}

// ---------------- router: one wave (32 lanes) per token ----------------
__global__ __launch_bounds__(256) void router_kernel(
    const float* __restrict__ x, const float* __restrict__ Wr,
    const float* __restrict__ br, int* __restrict__ topidx,
    float* __restrict__ topw, int* __restrict__ cnt_se)
{
    int lane = threadIdx.x & 31;
    int wave = threadIdx.x >> 5;
    int n = blockIdx.x * 8 + wave;
    if (n >= NT) return;

    float acc[NE];
#pragma unroll
    for (int e = 0; e < NE; ++e) acc[e] = 0.f;

    const float* xr = x + (size_t)n * DIM;
    for (int d = lane; d < DIM; d += 32) {
        float xv = xr[d];
        const float* wr = Wr + d * NE;
#pragma unroll
        for (int e = 0; e < NE; ++e) acc[e] += xv * wr[e];
    }
#pragma unroll
    for (int off = 16; off >= 1; off >>= 1) {
#pragma unroll
        for (int e = 0; e < NE; ++e) acc[e] += __shfl_xor(acc[e], off, 32);
    }
    if (lane == 0) {
        float lg[NE], m = -1e30f;
#pragma unroll
        for (int e = 0; e < NE; ++e) { lg[e] = acc[e] + br[e]; m = fmaxf(m, lg[e]); }
        float s = 0.f;
#pragma unroll
        for (int e = 0; e < NE; ++e) { lg[e] = __expf(lg[e] - m); s += lg[e]; }
        float inv = 1.f / s;
        // top-2, lowest index wins ties (strict >) to match jax top_k
        int i0 = -1; float b0 = -1e30f;
#pragma unroll
        for (int e = 0; e < NE; ++e) if (lg[e] > b0) { b0 = lg[e]; i0 = e; }
        int i1 = -1; float b1v = -1e30f;
#pragma unroll
        for (int e = 0; e < NE; ++e) if (e != i0 && lg[e] > b1v) { b1v = lg[e]; i1 = e; }
        topidx[n * 2 + 0] = i0;  topw[n * 2 + 0] = b0 * inv;
        topidx[n * 2 + 1] = i1;  topw[n * 2 + 1] = b1v * inv;
        atomicAdd(&cnt_se[i0], 1);
        atomicAdd(&cnt_se[8 + i1], 1);
    }
}

// ---------------- base offsets (64-aligned segments, +1 sentinel row) ----------------
__global__ void base_kernel(const int* __restrict__ cnt_se, int* __restrict__ base) {
    if (threadIdx.x == 0 && blockIdx.x == 0) {
        int b = 0;
        base[0] = 0;
        for (int e = 0; e < NE; ++e) {
            int c = cnt_se[e] + cnt_se[8 + e] + 1;  // tokens + sentinel
            b += ((c + 63) >> 6) << 6;
            base[e + 1] = b;
        }
    }
}

// ---------------- assign row positions ----------------
__global__ void assign_kernel(const int* __restrict__ topidx, const int* __restrict__ base,
                              int* __restrict__ fill, int* __restrict__ loc,
                              int* __restrict__ rowtok)
{
    int n = blockIdx.x * 256 + threadIdx.x;
    if (n >= NT) return;
#pragma unroll
    for (int s = 0; s < 2; ++s) {
        int e = topidx[n * 2 + s];
        int p = atomicAdd(&fill[e], 1);
        int r = base[e] + 1 + p;      // row 0 of each segment = sentinel (zeros)
        loc[n * 2 + s] = r;
        rowtok[r] = n;
    }
}

// ---------------- WMMA GEMM: 64 rows x 128 cols per block, 8 waves ----------------
// GATHER: rows come from x via rowtok (sentinel/pad rows -> zeros); else rows are dense.
// KDIM/NDIM compile-time so all B-matrix loads use immediate instruction offsets.
template <bool GATHER, bool RELU, int KDIM, int NDIM>
__global__ __launch_bounds__(256) void gemm_moe(
    const float* __restrict__ src, const float* __restrict__ W,
    const float* __restrict__ bias, float* __restrict__ dst,
    const int* __restrict__ basep, const int* __restrict__ rowtok)
{
    __shared__ float lds[64 * 132];          // 128-wide K chunk, padded stride

    int rt = blockIdx.y * 64;
    int total = basep[NE];
    if (rt >= total) return;
    int e = 0;
#pragma unroll
    for (int i = 1; i <= NE; ++i) if (rt >= basep[i]) e = i;   // e in [0,7]

    int tid  = threadIdx.x;
    int lane = tid & 31;
    int wave = tid >> 5;

    // loader mapping: 4 threads per row, 32 floats each
    int row_l = tid >> 2;
    int q     = tid & 3;
    int gr_ld = rt + row_l;
    bool valid;
    const float* srcrow;
    if (GATHER) {
        int t = rowtok[gr_ld];
        valid  = (t >= 0);
        srcrow = valid ? (src + (size_t)t * KDIM) : src;
    } else {
        valid  = true;
        srcrow = src + (size_t)gr_ld * KDIM;
    }
    srcrow += q * 32;

    v8f acc[4];
    const v8f vz = {0.f, 0.f, 0.f, 0.f, 0.f, 0.f, 0.f, 0.f};
#pragma unroll
    for (int r = 0; r < 4; ++r) acc[r] = vz;

    int col  = blockIdx.x * 128 + wave * 16 + (lane & 15);
    int koff = (lane >> 4) * 2;            // K sub-row pair select (A and B layouts match)
    int arow = lane & 15;

    // running pointer: covers K rows {koff, koff+1} of each K-step; all other
    // offsets are compile-time immediates (max 127*NDIM*4 bytes < 8MB ioffset range)
    const float* wk = W + (size_t)e * KDIM * NDIM + col + (size_t)koff * NDIM;
    // per-lane prefetch base: each lane prefetches 4 K-rows of next chunk
    const float* pf = W + (size_t)e * KDIM * NDIM + blockIdx.x * 128 + wave * 16
                        + (size_t)(lane * 4) * NDIM;

    constexpr int NCHUNK = KDIM >> 7;
#pragma unroll 1
    for (int kc = 0; kc < NCHUNK; ++kc) {
        __syncthreads();
        {
            const float* s = srcrow + kc * 128;
            float* l = &lds[row_l * 132 + q * 32];
#pragma unroll
            for (int i = 0; i < 8; ++i) {
                float4 v = valid ? *(const float4*)(s + i * 4)
                                 : make_float4(0.f, 0.f, 0.f, 0.f);
                *(float4*)(l + i * 4) = v;
            }
        }
        __syncthreads();

        if (kc + 1 < NCHUNK) {
            // pull next chunk of W toward the WGP (global_prefetch_b8)
            const float* p = pf + (size_t)((kc + 1) * 128) * NDIM;
#pragma unroll
            for (int i = 0; i < 4; ++i)
                __builtin_prefetch(p + (size_t)i * NDIM, 0, 0);
        }

#pragma unroll
        for (int ks = 0; ks < 32; ++ks) {
            v2f bfrag;
            bfrag.x = wk[(size_t)(ks * 4) * NDIM];
            bfrag.y = wk[(size_t)(ks * 4 + 1) * NDIM];
#pragma unroll
            for (int r = 0; r < 4; ++r) {
                const float* ap = &lds[(r * 16 + arow) * 132 + ks * 4 + koff];
                v2f afrag;
                afrag.x = ap[0];
                afrag.y = ap[1];
                acc[r] = __builtin_amdgcn_wmma_f32_16x16x4_f32(
                    false, afrag, false, bfrag, (short)0, acc[r], false, false);
            }
        }
        wk += (size_t)128 * NDIM;
    }

    float bv = bias[(size_t)e * NDIM + col];
    int mhalf = (lane >> 4) << 3;
#pragma unroll
    for (int r = 0; r < 4; ++r) {
#pragma unroll
        for (int v = 0; v < 8; ++v) {
            int rl = r * 16 + v + mhalf;
            float val = acc[r][v] + bv;
            if (RELU) val = fmaxf(val, 0.f);
            dst[(size_t)(rt + rl) * NDIM + col] = val;
        }
    }
}

// ---------------- C_s = sum over included experts of c_e (sentinel rows of y) ----------------
__global__ void cs_kernel(const float* __restrict__ y, const int* __restrict__ cnt_se,
                          const int* __restrict__ base, float* __restrict__ Cs)
{
    int d = blockIdx.x * 256 + threadIdx.x;
    if (d >= DIM) return;
    float a0 = 0.f, a1 = 0.f;
#pragma unroll
    for (int e = 0; e < NE; ++e) {
        float cv = y[(size_t)base[e] * DIM + d];
        if (cnt_se[e]     > 0) a0 += cv;
        if (cnt_se[8 + e] > 0) a1 += cv;
    }
    Cs[d]       = a0;
    Cs[DIM + d] = a1;
}

// ---------------- combine: out[n] = sum_s w_s*(y_loc - c_e + C_s) ----------------
__global__ __launch_bounds__(256) void combine_kernel(
    const float* __restrict__ y, const float* __restrict__ Cs,
    const int* __restrict__ loc, const int* __restrict__ topidx,
    const float* __restrict__ topw, const int* __restrict__ base,
    float* __restrict__ out)
{
    int n  = blockIdx.x;
    int d4 = threadIdx.x * 4;
    float w0 = topw[n * 2], w1 = topw[n * 2 + 1];
    int e0 = topidx[n * 2], e1 = topidx[n * 2 + 1];
    float4 ya = *(const float4*)(y + (size_t)loc[n * 2]     * DIM + d4);
    float4 yb = *(const float4*)(y + (size_t)loc[n * 2 + 1] * DIM + d4);
    float4 ca = *(const float4*)(y + (size_t)base[e0] * DIM + d4);
    float4 cb = *(const float4*)(y + (size_t)base[e1] * DIM + d4);
    float4 g0 = *(const float4*)(Cs + d4);
    float4 g1 = *(const float4*)(Cs + DIM + d4);
    float4 o;
    o.x = w0 * (ya.x - ca.x + g0.x) + w1 * (yb.x - cb.x + g1.x);
    o.y = w0 * (ya.y - ca.y + g0.y) + w1 * (yb.y - cb.y + g1.y);
    o.z = w0 * (ya.z - ca.z + g0.z) + w1 * (yb.z - cb.z + g1.z);
    o.w = w0 * (ya.w - ca.w + g0.w) + w1 * (yb.w - cb.w + g1.w);
    *(float4*)(out + (size_t)n * DIM + d4) = o;
}

// ---------------- host launch ----------------
extern "C" void kernel_launch(void* const* d_in, const int* in_sizes, int n_in,
                              void* d_out, int out_size, void* d_ws, size_t ws_size,
                              hipStream_t stream) {
    const float* x  = (const float*)d_in[0];
    const float* Wr = (const float*)d_in[1];
    const float* br = (const float*)d_in[2];
    const float* W1 = (const float*)d_in[3];
    const float* b1 = (const float*)d_in[4];
    const float* W2 = (const float*)d_in[5];
    const float* b2 = (const float*)d_in[6];
    // d_in[7] is k == 2 (compile-time here)
    float* out = (float*)d_out;

    char* ws = (char*)d_ws;
    int*   cnt_se = (int*)  (ws + OFF_CNT);
    int*   base   = (int*)  (ws + OFF_BASE);
    int*   fill   = (int*)  (ws + OFF_FILL);
    int*   topidx = (int*)  (ws + OFF_TOPIDX);
    float* topw   = (float*)(ws + OFF_TOPW);
    int*   loc    = (int*)  (ws + OFF_LOC);
    int*   rowtok = (int*)  (ws + OFF_ROWTOK);
    float* Cs     = (float*)(ws + OFF_CS);
    float* hbuf   = (float*)(ws + OFF_H);
    float* ybuf   = (float*)(ws + OFF_Y);

    init_kernel<<<(MAXROWS + 255) / 256, 256, 0, stream>>>(cnt_se, fill, rowtok);
    router_kernel<<<NT / 8, 256, 0, stream>>>(x, Wr, br, topidx, topw, cnt_se);
    base_kernel<<<1, 64, 0, stream>>>(cnt_se, base);
    assign_kernel<<<NT / 256, 256, 0, stream>>>(topidx, base, fill, loc, rowtok);

    // h = relu(gather(x) @ W1_e + b1_e)   [MAXROWS, HID]
    gemm_moe<true, true, DIM, HID><<<dim3(HID / 128, MAXROWS / 64), 256, 0, stream>>>(
        x, W1, b1, hbuf, base, rowtok);
    // y = h @ W2_e + b2_e                 [MAXROWS, DIM]
    gemm_moe<false, false, HID, DIM><<<dim3(DIM / 128, MAXROWS / 64), 256, 0, stream>>>(
        hbuf, W2, b2, ybuf, base, rowtok);

    cs_kernel<<<(DIM + 255) / 256, 256, 0, stream>>>(ybuf, cnt_se, base, Cs);
    combine_kernel<<<NT, 256, 0, stream>>>(ybuf, Cs, loc, topidx, topw, base, out);
}